// TransformerBlock_49074296324627
// MI455X (gfx1250) — compile-verified
//
#include <hip/hip_runtime.h>
#include <math.h>

// ---------------------------------------------------------------------------
// Fused tiny-transformer forward for MI455X (gfx1250), single persistent WGP.
// - All activations LDS-resident (one WGP, 320KB LDS), zero HBM intermediates.
// - Every matmul on V_WMMA_F32_16X16X4_F32 (exact fp32, matrix pipes).
// - Flash-style online softmax per head, wave32 shuffle reductions.
// - ds_add_f32 atomics for the fused multi-head output projection.
// - FFN weights staged per-layer via GLOBAL_LOAD_ASYNC_TO_LDS (ASYNCcnt),
//   issued at layer start and consumed after attention -> full overlap.
// ---------------------------------------------------------------------------

typedef float v2f __attribute__((ext_vector_type(2)));
typedef float v8f __attribute__((ext_vector_type(8)));

#define NTHREADS 256
#define NWAVE 8

// LDS layout (float offsets). SP = 112 padded rows, D = 32.
//   TOK   [112][32]               @ 0
//   ACC   [112][32]               @ 3584
//   KBUF  8 x [112][32]           @ 7168
//   VBUF  8 x [112][32]           @ 35840
//   WS    8 x (QT/OT 512 | PT 256)@ 64512   (OT aliases QT: q dead before o)
//   FW1S  [32][128]               @ 70656   (async-staged fW1 panel)
//   FW2S  [128][32]               @ 74752   (async-staged fW2 panel)
//   FFH   [112][128] aliases KBUF; FFO [112][32] aliases KBUF+14336
#define SMEM_FLOATS 78848 // 315392 B < 320 KB
#define FW1_LDS_BYTE 282624u // 70656*4
#define FW2_LDS_BYTE 299008u // 74752*4

struct Args {
  const float *x, *pW, *pb, *pls, *plb;
  const float *faW1, *fab1, *faW2, *fab2;
  const float *peW1, *peb1, *peW2, *peb2, *pe_sw, *pe_lw, *pe_le;
  const float *cW, *cb, *cls, *clb;
  const float *Wq, *bq, *Wk, *bk, *Wv, *bv, *Wo, *bo;
  const float *fW1, *fb1, *fW2, *fb2, *l1s, *l1b, *l2s, *l2b;
  float *out;
};

// ---- exact-fp32 WMMA: D(16x16) = A(16x4) * B(4x16) + C -------------------
__device__ __forceinline__ v8f wmma4(v2f a, v2f b, v8f c) {
  return __builtin_amdgcn_wmma_f32_16x16x4_f32(false, a, false, b, (short)0, c,
                                               false, false);
}

// A fragment 16x4 (row-major source, ld = row stride)
__device__ __forceinline__ v2f ldA(const float* p, int ld) {
  int l = threadIdx.x & 31;
  int r = l & 15, kk = (l >> 4) << 1;
  v2f a;
  a.x = p[r * ld + kk];
  a.y = p[r * ld + kk + 1];
  return a;
}
// B fragment 4x16 (row-major source)
__device__ __forceinline__ v2f ldB(const float* p, int ld) {
  int l = threadIdx.x & 31;
  int c = l & 15, kk = (l >> 4) << 1;
  v2f b;
  b.x = p[kk * ld + c];
  b.y = p[(kk + 1) * ld + c];
  return b;
}
// B fragment from a transposed source: B[k][c] = p[c*ld + k]  (for Q @ K^T)
__device__ __forceinline__ v2f ldBT(const float* p, int ld) {
  int l = threadIdx.x & 31;
  int c = l & 15, kk = (l >> 4) << 1;
  v2f b;
  b.x = p[c * ld + kk];
  b.y = p[c * ld + kk + 1];
  return b;
}
// store C/D fragment: VGPR e -> row r0+e (r0 = (lane>=16)*8), col = lane&15
__device__ __forceinline__ void stD(float* p, int ld, v8f d) {
  int l = threadIdx.x & 31;
  int c = l & 15, r0 = (l >> 4) << 3;
#pragma unroll
  for (int e = 0; e < 8; ++e) p[(r0 + e) * ld + c] = d[e];
}

template <int K>
__device__ __forceinline__ v8f gemmTile(const float* A, int lda, const float* B,
                                        int ldb, v8f acc) {
#pragma unroll
  for (int k = 0; k < K; k += 4)
    acc = wmma4(ldA(A + k, lda), ldB(B + k * ldb, ldb), acc);
  return acc;
}

// ---- CDNA5 async DMA: global -> LDS, tracked with ASYNCcnt ----------------
__device__ __forceinline__ void asyncLds16(unsigned ldsByte, const void* g) {
  asm volatile("global_load_async_to_lds_b128 %0, %1, off"
               :
               : "v"(ldsByte), "v"(g)
               : "memory");
}
__device__ __forceinline__ void waitAsync0() {
  asm volatile("s_wait_asynccnt 0" ::: "memory");
}

// ---- wave32 reductions ----------------------------------------------------
__device__ __forceinline__ float rmax16(float x) {
#pragma unroll
  for (int m = 1; m < 16; m <<= 1) x = fmaxf(x, __shfl_xor(x, m, 32));
  return x;
}
__device__ __forceinline__ float rsum16(float x) {
#pragma unroll
  for (int m = 1; m < 16; m <<= 1) x += __shfl_xor(x, m, 32);
  return x;
}
__device__ __forceinline__ float rsum32(float x) {
#pragma unroll
  for (int m = 1; m < 32; m <<= 1) x += __shfl_xor(x, m, 32);
  return x;
}
__device__ __forceinline__ float rmax32(float x) {
#pragma unroll
  for (int m = 1; m < 32; m <<= 1) x = fmaxf(x, __shfl_xor(x, m, 32));
  return x;
}

__global__ __launch_bounds__(NTHREADS) void fused_transformer(Args p) {
  extern __shared__ float sm[];
  float* TOK = sm;            // [112][32]
  float* ACC = sm + 3584;     // [112][32]
  float* KB = sm + 7168;      // 8 x [112][32]
  float* VB = sm + 35840;     // 8 x [112][32]
  float* WS = sm + 64512;     // 8 x 768
  float* FW1S = sm + 70656;   // [32][128]
  float* FW2S = sm + 74752;   // [128][32]
  float* FFH = sm + 7168;     // [112][128] (aliases KB, FFN phase only)
  float* FFO = sm + 21504;    // [112][32]  (aliases KB region)

  const int tid = threadIdx.x;
  const int lane = tid & 31;
  const int wv = tid >> 5;

  // ---- stage x into LDS, zero pad rows 100..111 ------------------------
  for (int i = tid; i < 3584; i += NTHREADS) TOK[i] = (i < 3200) ? p.x[i] : 0.f;
  __syncthreads();

  // ---- Phase A: tok = gelu(LN(x@pW + pb)) + x  (one wave per row) ------
  for (int s = wv; s < 100; s += NWAVE) {
    float acc = p.pb[lane];
    for (int k = 0; k < 32; ++k) acc += TOK[s * 32 + k] * p.pW[k * 32 + lane];
    float mn = rsum32(acc) * 0.03125f;
    float d = acc - mn;
    float var = rsum32(d * d) * 0.03125f;
    float h = d * rsqrtf(var + 1e-5f) * p.pls[lane] + p.plb[lane];
    float g = 0.5f * h * (1.f + erff(h * 0.70710678118f));
    TOK[s * 32 + lane] = g + TOK[s * 32 + lane];
  }
  __syncthreads();

  // ---- Phase B: feature-attention gate ---------------------------------
  for (int s = wv; s < 100; s += NWAVE) {
    float hid = 0.f;
    if (lane < 16) {
      float a = p.fab1[lane];
      for (int k = 0; k < 32; ++k) a += TOK[s * 32 + k] * p.faW1[k * 16 + lane];
      hid = fmaxf(a, 0.f);
    }
    float logit = p.fab2[lane];
    for (int j = 0; j < 16; ++j)
      logit += __shfl(hid, j, 32) * p.faW2[j * 32 + lane];
    float mx = rmax32(logit);
    float e = __expf(logit - mx);
    float sum = rsum32(e);
    TOK[s * 32 + lane] *= e / sum;
  }
  __syncthreads();

  // ---- Phase C: adaptive positional encoding ---------------------------
  for (int s = wv; s < 100; s += NWAVE) {
    float contrib = 0.f;
    if (lane < 8) {
      float a = p.peb1[lane];
      for (int k = 0; k < 32; ++k) a += TOK[s * 32 + k] * p.peW1[k * 8 + lane];
      contrib = fmaxf(a, 0.f) * p.peW2[lane];
    }
    float z = rsum32(contrib) + p.peb2[0];
    float pw = 1.f / (1.f + __expf(-z));
    int j = lane >> 1;
    float div = __expf((float)(2 * j) * -0.28782313662f); // -ln(1e4)/32
    float ang = (float)s / div;
    float enc = (lane & 1) ? __cosf(ang) : __sinf(ang);
    TOK[s * 32 + lane] +=
        (enc * p.pe_sw[0] + p.pe_le[s * 32 + lane] * p.pe_lw[0]) * pw;
  }
  __syncthreads();

  // ---- Phase D: conv1d(k=3,pad=1) residual + LN ------------------------
  for (int s = wv; s < 100; s += NWAVE) {
    float a = p.cb[lane];
    for (int ic = 0; ic < 32; ++ic) {
      float w0 = p.cW[(lane * 32 + ic) * 3 + 0];
      float w1 = p.cW[(lane * 32 + ic) * 3 + 1];
      float w2 = p.cW[(lane * 32 + ic) * 3 + 2];
      float xm = (s > 0) ? TOK[(s - 1) * 32 + ic] : 0.f;
      float x0 = TOK[s * 32 + ic];
      float xp = (s < 99) ? TOK[(s + 1) * 32 + ic] : 0.f;
      a += w0 * xm + w1 * x0 + w2 * xp;
    }
    FFO[s * 32 + lane] = TOK[s * 32 + lane] + fmaxf(a, 0.f);
  }
  __syncthreads();
  for (int s = wv; s < 100; s += NWAVE) {
    float y = FFO[s * 32 + lane];
    float mn = rsum32(y) * 0.03125f;
    float d = y - mn;
    float var = rsum32(d * d) * 0.03125f;
    TOK[s * 32 + lane] = d * rsqrtf(var + 1e-5f) * p.cls[lane] + p.clb[lane];
  }
  __syncthreads();

  // ---- 12 transformer layers; wave wv owns head wv ---------------------
  float* QT = WS + wv * 768; // 16x32 q tile
  float* PT = QT + 512;      // 16x16 probability tile
  float* OT = QT;            // 16x32 o tile (aliases QT: q dead by then)
  float* KH = KB + wv * 3584;
  float* VH = VB + wv * 3584;

  for (int li = 0; li < 12; ++li) {
    // Kick off async DMA of this layer's FFN weight panels into LDS.
    // Consumed only after attention -> the whole attention phase hides it.
    {
      const char* g1 = (const char*)(p.fW1 + (size_t)li * 4096);
      const char* g2 = (const char*)(p.fW2 + (size_t)li * 4096);
#pragma unroll
      for (int c = 0; c < 4; ++c) {
        unsigned off = (unsigned)(tid + c * 256) * 16u;
        asyncLds16(FW1_LDS_BYTE + off, g1 + off);
        asyncLds16(FW2_LDS_BYTE + off, g2 + off);
      }
    }
    if (li + 1 < 12) { // warm L2 for next layer's attention weights
      __builtin_prefetch((const char*)(p.Wq + (size_t)(li + 1) * 8192) +
                             tid * 128, 0, 1);
    }
    for (int i = tid; i < 3584; i += NTHREADS) ACC[i] = 0.f;
    __syncthreads();

    // K,V panels for this head (producer == consumer wave; no sync needed)
    const float* WkL = p.Wk + (size_t)(li * 8 + wv) * 1024;
    const float* WvL = p.Wv + (size_t)(li * 8 + wv) * 1024;
    for (int ri = 0; ri < 7; ++ri) {
      for (int cj = 0; cj < 2; ++cj) {
        v8f ka = {};
        ka = gemmTile<32>(TOK + ri * 16 * 32, 32, WkL + cj * 16, 32, ka);
        float bkv = p.bk[(li * 8 + wv) * 32 + cj * 16 + (lane & 15)];
#pragma unroll
        for (int e = 0; e < 8; ++e) ka[e] += bkv;
        stD(KH + ri * 16 * 32 + cj * 16, 32, ka);

        v8f va = {};
        va = gemmTile<32>(TOK + ri * 16 * 32, 32, WvL + cj * 16, 32, va);
        float bvv = p.bv[(li * 8 + wv) * 32 + cj * 16 + (lane & 15)];
#pragma unroll
        for (int e = 0; e < 8; ++e) va[e] += bvv;
        stD(VH + ri * 16 * 32 + cj * 16, 32, va);
      }
    }

    // Flash-style causal attention + fused per-head Wo projection
    const float* WqL = p.Wq + (size_t)(li * 8 + wv) * 1024;
    const float* WoL = p.Wo + (size_t)li * 256 * 32 + wv * 32 * 32;
    const int cbase = lane & 15;
    for (int qi = 0; qi < 7; ++qi) {
      for (int cj = 0; cj < 2; ++cj) { // q tile -> QT
        v8f qa = {};
        qa = gemmTile<32>(TOK + qi * 16 * 32, 32, WqL + cj * 16, 32, qa);
        float bqv = p.bq[(li * 8 + wv) * 32 + cj * 16 + (lane & 15)];
#pragma unroll
        for (int e = 0; e < 8; ++e) qa[e] += bqv;
        stD(QT + cj * 16, 32, qa);
      }

      float m_[8], l_[8];
#pragma unroll
      for (int e = 0; e < 8; ++e) { m_[e] = -3.0e38f; l_[e] = 0.f; }
      v8f o0 = {}, o1 = {};
      const int rbase = qi * 16 + ((lane >> 4) << 3);

      for (int kj = 0; kj <= qi; ++kj) { // causal: only lower tiles
        v8f sc = {};
#pragma unroll
        for (int kk = 0; kk < 32; kk += 4)
          sc = wmma4(ldA(QT + kk, 32), ldBT(KH + kj * 16 * 32 + kk, 32), sc);
        const int c = kj * 16 + cbase;
#pragma unroll
        for (int e = 0; e < 8; ++e) { // scale + causal/pad mask on D-frag
          float sv = sc[e] * 0.17677669529f; // 1/sqrt(32)
          int r = rbase + e;
          sc[e] = (c <= r && c < 100) ? sv : -1e30f;
        }
        float alpha[8];
#pragma unroll
        for (int e = 0; e < 8; ++e) { // online softmax, rows in 16-lane groups
          float tm = rmax16(sc[e]);
          float nm = fmaxf(m_[e], tm);
          float al = __expf(m_[e] - nm);
          float pv = __expf(sc[e] - nm);
          float rs = rsum16(pv);
          l_[e] = l_[e] * al + rs;
          m_[e] = nm;
          sc[e] = pv;
          alpha[e] = al;
        }
#pragma unroll
        for (int e = 0; e < 8; ++e) { o0[e] *= alpha[e]; o1[e] *= alpha[e]; }
        stD(PT, 16, sc); // re-fragment P as an A operand via LDS
        const float* Vt = VH + kj * 16 * 32;
#pragma unroll
        for (int kk = 0; kk < 16; kk += 4) {
          v2f a = ldA(PT + kk, 16);
          o0 = wmma4(a, ldB(Vt + kk * 32, 32), o0);
          o1 = wmma4(a, ldB(Vt + kk * 32 + 16, 32), o1);
        }
      }
#pragma unroll
      for (int e = 0; e < 8; ++e) {
        float inv = 1.f / l_[e];
        o0[e] *= inv;
        o1[e] *= inv;
      }
      stD(OT, 32, o0); // OT aliases QT (q fully consumed)
      stD(OT + 16, 32, o1);
      for (int cj = 0; cj < 2; ++cj) { // o @ Wo[head block] -> ACC (atomic)
        v8f pa = {};
        pa = gemmTile<32>(OT, 32, WoL + cj * 16, 32, pa);
        int c = cj * 16 + cbase;
        int r0 = qi * 16 + ((lane >> 4) << 3);
#pragma unroll
        for (int e = 0; e < 8; ++e) atomicAdd(&ACC[(r0 + e) * 32 + c], pa[e]);
      }
    }
    __syncthreads();

    // LN1: a = LN(ACC + bo + tok) -> TOK
    for (int s = wv; s < 100; s += NWAVE) {
      float y = ACC[s * 32 + lane] + p.bo[li * 32 + lane] + TOK[s * 32 + lane];
      float mn = rsum32(y) * 0.03125f;
      float d = y - mn;
      float var = rsum32(d * d) * 0.03125f;
      TOK[s * 32 + lane] =
          d * rsqrtf(var + 1e-5f) * p.l1s[li * 32 + lane] + p.l1b[li * 32 + lane];
    }
    waitAsync0(); // FFN weight panels landed in LDS (issued at layer start)
    __syncthreads();

    // FFN1: FFH = relu(a @ fW1 + fb1)  (B from LDS panel; 56 tiles / 8 waves)
    for (int t = wv; t < 56; t += NWAVE) {
      int ri = t >> 3, cj = t & 7;
      v8f a = {};
      a = gemmTile<32>(TOK + ri * 16 * 32, 32, FW1S + cj * 16, 128, a);
      float b1 = p.fb1[li * 128 + cj * 16 + (lane & 15)];
#pragma unroll
      for (int e = 0; e < 8; ++e) a[e] = fmaxf(a[e] + b1, 0.f);
      stD(FFH + ri * 16 * 128 + cj * 16, 128, a);
    }
    __syncthreads();

    // FFN2: FFO = FFH @ fW2 + fb2 + a  (B from LDS panel; 14 tiles)
    for (int t = wv; t < 14; t += NWAVE) {
      int ri = t >> 1, cj = t & 1;
      v8f a = {};
      a = gemmTile<128>(FFH + ri * 16 * 128, 128, FW2S + cj * 16, 32, a);
      float b2 = p.fb2[li * 32 + cj * 16 + (lane & 15)];
      int c = cj * 16 + (lane & 15);
      int r0 = ri * 16 + ((lane >> 4) << 3);
#pragma unroll
      for (int e = 0; e < 8; ++e) a[e] += b2 + TOK[(r0 + e) * 32 + c];
      stD(FFO + ri * 16 * 32 + cj * 16, 32, a);
    }
    __syncthreads();

    // LN2 -> TOK (next layer input)
    for (int s = wv; s < 100; s += NWAVE) {
      float y = FFO[s * 32 + lane];
      float mn = rsum32(y) * 0.03125f;
      float d = y - mn;
      float var = rsum32(d * d) * 0.03125f;
      TOK[s * 32 + lane] =
          d * rsqrtf(var + 1e-5f) * p.l2s[li * 32 + lane] + p.l2b[li * 32 + lane];
    }
    __syncthreads();
  }

  for (int i = tid; i < 3200; i += NTHREADS) p.out[i] = TOK[i];
}

extern "C" void kernel_launch(void* const* d_in, const int* in_sizes, int n_in,
                              void* d_out, int out_size, void* d_ws,
                              size_t ws_size, hipStream_t stream) {
  (void)in_sizes; (void)n_in; (void)out_size; (void)d_ws; (void)ws_size;
  Args a;
  a.x = (const float*)d_in[0];
  a.pW = (const float*)d_in[1];   a.pb = (const float*)d_in[2];
  a.pls = (const float*)d_in[3];  a.plb = (const float*)d_in[4];
  a.faW1 = (const float*)d_in[5]; a.fab1 = (const float*)d_in[6];
  a.faW2 = (const float*)d_in[7]; a.fab2 = (const float*)d_in[8];
  a.peW1 = (const float*)d_in[9]; a.peb1 = (const float*)d_in[10];
  a.peW2 = (const float*)d_in[11]; a.peb2 = (const float*)d_in[12];
  a.pe_sw = (const float*)d_in[13]; a.pe_lw = (const float*)d_in[14];
  a.pe_le = (const float*)d_in[15];
  a.cW = (const float*)d_in[16];  a.cb = (const float*)d_in[17];
  a.cls = (const float*)d_in[18]; a.clb = (const float*)d_in[19];
  a.Wq = (const float*)d_in[20];  a.bq = (const float*)d_in[21];
  a.Wk = (const float*)d_in[22];  a.bk = (const float*)d_in[23];
  a.Wv = (const float*)d_in[24];  a.bv = (const float*)d_in[25];
  a.Wo = (const float*)d_in[26];  a.bo = (const float*)d_in[27];
  a.fW1 = (const float*)d_in[28]; a.fb1 = (const float*)d_in[29];
  a.fW2 = (const float*)d_in[30]; a.fb2 = (const float*)d_in[31];
  a.l1s = (const float*)d_in[32]; a.l1b = (const float*)d_in[33];
  a.l2s = (const float*)d_in[34]; a.l2b = (const float*)d_in[35];
  a.out = (float*)d_out;

  const size_t smem = (size_t)SMEM_FLOATS * sizeof(float); // ~308 KB of 320 KB
  hipFuncSetAttribute((const void*)fused_transformer,
                      hipFuncAttributeMaxDynamicSharedMemorySize, (int)smem);
  fused_transformer<<<dim3(1), dim3(NTHREADS), smem, stream>>>(a);
}